// STPNet_58995670778380
// MI455X (gfx1250) — compile-verified
//
#include <hip/hip_runtime.h>

#define B_   128
#define S_   8192
#define D_   64
#define H_   16
#define NC_  64      // chunks per sequence
#define CL_  128     // chunk length (NC_*CL_ == S_)
#define INV_TAU (1.0f/6.0f)
#define SYN_U   0.5f

typedef float v2f __attribute__((ext_vector_type(2)));
typedef float v8f __attribute__((ext_vector_type(8)));

// ---------------------------------------------------------------------------
// Pass 1: per (b, chunk, d) compose the chunk's affine map  syn_out = A*syn_in + C
//   step t:  syn' = e_t*syn + b_t,  e_t = exp(-k_t), b_t = (1/tau)*(1-e_t)/k_t
// Lane -> d mapping gives fully coalesced 128B loads per wave.
// ---------------------------------------------------------------------------
__global__ void stp_chunk_compose(const float* __restrict__ x,
                                  float* __restrict__ Aarr,
                                  float* __restrict__ Carr) {
    int tid = blockIdx.x * blockDim.x + threadIdx.x;   // B_*NC_*D_ threads
    int d = tid & (D_ - 1);
    int c = (tid >> 6) & (NC_ - 1);
    int b = tid >> 12;
    const float* xp = x + ((size_t)b * S_ + (size_t)c * CL_) * D_ + d;
    float A = 1.0f, C = 0.0f;
    for (int t = 0; t < CL_; ++t) {
        float k = INV_TAU + SYN_U * xp[(size_t)t * D_];
        float e = __expf(-k);
        float bb = INV_TAU * (1.0f - e) / k;
        A = e * A;
        C = e * C + bb;
    }
    int idx = (b * NC_ + c) * D_ + d;
    Aarr[idx] = A;
    Carr[idx] = C;
}

// ---------------------------------------------------------------------------
// Pass 2: per (b,d) serial prefix over the 64 chunk summaries -> start state
//         of every chunk.  Tiny kernel (8192 threads x 64 steps).
// ---------------------------------------------------------------------------
__global__ void stp_chunk_scan(const float* __restrict__ syn0,
                               const float* __restrict__ Aarr,
                               const float* __restrict__ Carr,
                               float* __restrict__ startArr) {
    int tid = blockIdx.x * blockDim.x + threadIdx.x;   // B_*D_ threads
    int d = tid & (D_ - 1);
    int b = tid >> 6;
    float syn = syn0[b * D_ + d];
    for (int c = 0; c < NC_; ++c) {
        int idx = (b * NC_ + c) * D_ + d;
        startArr[idx] = syn;
        syn = Aarr[idx] * syn + Carr[idx];
    }
}

// ---------------------------------------------------------------------------
// Pass 3: replay each chunk from its start state; emit input_syn (re-read by
// the GEMM -> regular stores) and the verbatim inputs copy (never re-read ->
// non-temporal stores).
// ---------------------------------------------------------------------------
__global__ void stp_chunk_emit(const float* __restrict__ x,
                               const float* __restrict__ startArr,
                               float* __restrict__ syn_out,
                               float* __restrict__ x_copy) {
    int tid = blockIdx.x * blockDim.x + threadIdx.x;   // B_*NC_*D_ threads
    int d = tid & (D_ - 1);
    int c = (tid >> 6) & (NC_ - 1);
    int b = tid >> 12;
    size_t base = ((size_t)b * S_ + (size_t)c * CL_) * D_ + d;
    float syn = startArr[(b * NC_ + c) * D_ + d];
    for (int t = 0; t < CL_; ++t) {
        size_t off = base + (size_t)t * D_;
        float xv = x[off];
        syn_out[off] = syn;                         // state BEFORE step t
        __builtin_nontemporal_store(xv, x_copy + off);
        float k = INV_TAU + SYN_U * xv;
        float e = __expf(-k);
        syn = e * syn + INV_TAU * (1.0f - e) / k;
    }
}

// ---------------------------------------------------------------------------
// Fused MLP: hidden = relu((syn*x) @ W1^T + b1) ; out = hidden @ W2^T + b2
// One wave per 16-row tile; K=64 as 16 x v_wmma_f32_16x16x4_f32; N=16 = H.
// A-frag (16x4 f32): lanes 0-15 -> M=lane, K {0,1}; lanes 16-31 -> M=lane-16,
// K {2,3}  => contiguous float2 per lane.  B-frags (W1^T) hoisted in VGPRs.
// ---------------------------------------------------------------------------
__global__ void stp_mlp_wmma(const float* __restrict__ x,
                             const float* __restrict__ syn,
                             const float* __restrict__ W1,
                             const float* __restrict__ b1,
                             const float* __restrict__ W2,
                             const float* __restrict__ b2,
                             float* __restrict__ hidden,
                             float* __restrict__ out,
                             int nrows) {
    int lane = threadIdx.x & 31;
    int wavesPerBlock = blockDim.x >> 5;
    int wave = blockIdx.x * wavesPerBlock + (threadIdx.x >> 5);
    int nwaves = gridDim.x * wavesPerBlock;

    int n    = lane & 15;            // N index (and A-frag row)
    int koff = (lane >> 4) << 1;     // 0 for lanes 0-15, 2 for lanes 16-31

    // Preload all 16 B fragments of W1^T:  B[k, n] = W1[n*64 + k]
    v2f bfrag[16];
#pragma unroll
    for (int s = 0; s < 16; ++s)
        bfrag[s] = *(const v2f*)(W1 + n * 64 + s * 4 + koff);

    float bias1 = b1[n];
    float w2v   = W2[n];
    float bias2 = b2[0];

    int ntiles = nrows >> 4;
    for (int tile = wave; tile < ntiles; tile += nwaves) {
        size_t row0 = (size_t)tile << 4;
        const float* xr = x   + (row0 + n) * 64 + koff;
        const float* sr = syn + (row0 + n) * 64 + koff;
        v8f acc = {};
#pragma unroll
        for (int s = 0; s < 16; ++s) {
            v2f xa = *(const v2f*)(xr + s * 4);
            v2f sa = *(const v2f*)(sr + s * 4);
            v2f a  = xa * sa;                         // elementwise syn * x
            acc = __builtin_amdgcn_wmma_f32_16x16x4_f32(
                      false, a, false, bfrag[s], (short)0, acc, false, false);
        }
        int rbase = (lane >> 4) << 3;                 // 0 or 8 (C/D layout)
#pragma unroll
        for (int j = 0; j < 8; ++j) {
            float h = acc[j] + bias1;
            h = fmaxf(h, 0.0f);                       // relu
            size_t r = row0 + rbase + j;
            __builtin_nontemporal_store(h, hidden + r * H_ + n);
            float o = h * w2v;                        // H=16 dot via shuffles
            o += __shfl_xor(o, 1, 32);
            o += __shfl_xor(o, 2, 32);
            o += __shfl_xor(o, 4, 32);
            o += __shfl_xor(o, 8, 32);
            if (n == 0)
                __builtin_nontemporal_store(o + bias2, out + r);
        }
    }
}

// ---------------------------------------------------------------------------
extern "C" void kernel_launch(void* const* d_in, const int* in_sizes, int n_in,
                              void* d_out, int out_size, void* d_ws, size_t ws_size,
                              hipStream_t stream) {
    (void)in_sizes; (void)n_in; (void)out_size; (void)ws_size;
    const float* inputs = (const float*)d_in[0];
    const float* syn_x  = (const float*)d_in[1];
    const float* W1     = (const float*)d_in[2];
    const float* b1     = (const float*)d_in[3];
    const float* W2     = (const float*)d_in[4];
    const float* b2     = (const float*)d_in[5];

    const size_t BS = (size_t)B_ * S_;
    float* out_y   = (float*)d_out;            // [B,S,1]
    float* out_hid = out_y + BS;               // [B,S,16]
    float* out_x   = out_y + BS * 17;          // [B,S,64]  (inputs copy)
    float* out_syn = out_y + BS * 81;          // [B,S,64]  (input_syn)

    const size_t NSUM = (size_t)B_ * D_ * NC_; // 512K chunk summaries
    float* Aarr  = (float*)d_ws;               // 2 MB
    float* Carr  = Aarr + NSUM;                // 2 MB
    float* start = Carr + NSUM;                // 2 MB

    stp_chunk_compose<<<(B_ * D_ * NC_) / 256, 256, 0, stream>>>(inputs, Aarr, Carr);
    stp_chunk_scan<<<(B_ * D_) / 256, 256, 0, stream>>>(syn_x, Aarr, Carr, start);
    stp_chunk_emit<<<(B_ * D_ * NC_) / 256, 256, 0, stream>>>(inputs, start, out_syn, out_x);
    stp_mlp_wmma<<<1024, 256, 0, stream>>>(inputs, out_syn, W1, b1, W2, b2,
                                           out_hid, out_y, (int)BS);
}